// Net_70798240907731
// MI455X (gfx1250) — compile-verified
//
#include <hip/hip_runtime.h>
#include <hip/hip_bf16.h>

// Fused conv3x3(same) -> maxpool2x2 -> conv3x3(same) -> maxpool2x2
// x: [2,1,4096,4096] f32, W1/W2: [1,1,3,3] f32, out: [2,1,1024,1024] f32.
// One workgroup (128 threads = 4 wave32) -> one 16x16 output tile.
// Conv2 is computed as banded-matrix WMMA (v_wmma_f32_16x16x4_f32).
// Interior x-tiles are staged with CDNA5 async Global->LDS B128 copies
// (ASYNCcnt pipe), border tiles with guarded scalar loads (zero padding).

typedef __attribute__((ext_vector_type(2))) float v2f;
typedef __attribute__((ext_vector_type(8))) float v8f;
typedef __attribute__((ext_vector_type(4))) int   v4i;

#define H_IN  4096
#define W_IN  4096
#define H2DIM 2048   // h2 spatial size (H_IN/2)
#define HOUT  1024   // output spatial size (H_IN/4)

#define XS_S 72      // x-tile LDS row stride (70 cols used, 72 = 18 float4)
#define H2_S 36      // h2 LDS row stride (34 used + 2 zero pad rows for K padding)
#define H3_S 33      // h3 LDS row stride (32 used, odd stride vs 64 banks)

#if defined(__gfx1250__) && \
    __has_builtin(__builtin_amdgcn_global_load_async_to_lds_b128) && \
    __has_builtin(__builtin_amdgcn_s_wait_asynccnt)
#define USE_ASYNC_LDS 1
#else
#define USE_ASYNC_LDS 0
#endif

#if USE_ASYNC_LDS
typedef v4i __attribute__((address_space(1)))* gv4i_p;  // global (AS1) int4*
typedef v4i __attribute__((address_space(3)))* lv4i_p;  // LDS (AS3) int4*

__device__ __forceinline__ void async_copy16(const float* gsrc, float* ldst) {
  // global_load_async_to_lds_b128: LDS written directly, no VGPR round-trip.
  __builtin_amdgcn_global_load_async_to_lds_b128(
      (gv4i_p)gsrc, (lv4i_p)ldst, /*offset=*/0, /*cpol=*/0);
}
#endif

__device__ __forceinline__ float tap3(int t, float k0, float k1, float k2) {
  // banded-matrix coefficient: W2 column tap if t in {0,1,2}, else 0
  float r = 0.0f;
  r = (t == 0) ? k0 : r;
  r = (t == 1) ? k1 : r;
  r = (t == 2) ? k2 : r;
  return r;
}

__global__ __launch_bounds__(128)
void fused_conv_pool_kernel(const float* __restrict__ x,
                            const float* __restrict__ W1,
                            const float* __restrict__ W2,
                            float* __restrict__ out) {
  __shared__ float xs[70 * XS_S];     // 70x70 input tile (zero-padded at image edge)
  __shared__ float h2s[H2_S * H2_S];  // 34x34 pooled-conv1 tile (+zero pad)
  __shared__ float h3s[32 * H3_S];    // 32x32 conv2 result before pool2

  const int tid = threadIdx.x;
  const int n   = blockIdx.z;
  const int oh0 = blockIdx.y * 16;    // output tile origin
  const int ow0 = blockIdx.x * 16;
  const int xr0 = 4 * oh0 - 3;        // x-tile origin (includes both conv halos)
  const int xc0 = 4 * ow0 - 3;

  // Broadcast 3x3 kernels into registers (scalar-cached loads).
  float w1v[9], w2v[9];
  #pragma unroll
  for (int t = 0; t < 9; ++t) { w1v[t] = W1[t]; w2v[t] = W2[t]; }

  // ---- Phase A: zero h2 staging, load 70x70 x tile into LDS ----
  for (int idx = tid; idx < H2_S * H2_S; idx += 128) h2s[idx] = 0.0f;
  const float* xbase = x + (size_t)n * H_IN * W_IN;
  // Strict in-bounds test covering the full 70x72 staged footprint.
  const bool interior = (xr0 >= 0) && (xr0 + 70 <= H_IN) &&
                        (xc0 >= 0) && (xc0 + XS_S <= W_IN);
#if USE_ASYNC_LDS
  if (interior) {
    // Async Global->LDS: 70 rows x 72 floats = 1260 x b128, ~10 per lane.
    const float* tbase = xbase + (size_t)xr0 * W_IN + xc0;
    for (int f4 = tid; f4 < (70 * XS_S) / 4; f4 += 128) {
      int e = f4 * 4;
      int r = e / XS_S, c = e - XS_S * r;   // float4 never crosses an LDS row
      async_copy16(tbase + (size_t)r * W_IN + c, &xs[e]);
    }
    __builtin_amdgcn_s_wait_asynccnt(0);
  } else
#else
  (void)interior;
#endif
  {
    for (int idx = tid; idx < 70 * 70; idx += 128) {
      int r = idx / 70, c = idx - 70 * r;
      int gr = xr0 + r, gc = xc0 + c;
      float v = 0.0f;
      if ((unsigned)gr < (unsigned)H_IN && (unsigned)gc < (unsigned)W_IN)
        v = xbase[(size_t)gr * W_IN + gc];
      xs[r * XS_S + c] = v;
    }
  }
  __syncthreads();

  // ---- Phase B: conv1 + pool1 -> 34x34 h2 tile (VALU FMA) ----
  for (int idx = tid; idx < 34 * 34; idx += 128) {
    int i = idx / 34, j = idx - 34 * i;
    int gi = 2 * oh0 - 1 + i;                 // h2 global coords (conv2 halo)
    int gj = 2 * ow0 - 1 + j;
    if ((unsigned)gi < (unsigned)H2DIM && (unsigned)gj < (unsigned)H2DIM) {
      float m = -3.402823466e+38f;
      #pragma unroll
      for (int a = 0; a < 2; ++a) {
        #pragma unroll
        for (int b = 0; b < 2; ++b) {
          int r = 2 * i + a, c = 2 * j + b;   // h1-local; taps xs rows r..r+2
          float s = 0.0f;
          #pragma unroll
          for (int dh = 0; dh < 3; ++dh)
            #pragma unroll
            for (int dw = 0; dw < 3; ++dw)
              s = fmaf(w1v[3 * dh + dw], xs[(r + dh) * XS_S + (c + dw)], s);
          m = fmaxf(m, s);
        }
      }
      h2s[i * H2_S + j] = m;
    } // else: stays zero (conv2 'same' zero padding)
  }
  __syncthreads();

  // ---- Phase C: conv2 as 3 column-shifted banded matmuls on WMMA ----
  // Each wave computes one 16x16 tile of the 32x32 h3.
  // D[M,N] = sum_{dw=0..2} sum_{r=0..17} A_dw[M,r] * h2s[16*th+r][16*tw+N+dw]
  // A_dw[M,r] = W2[r-M][dw] if 0<=r-M<=2 else 0; K=18 padded to 20 (rows 18,19
  // hit the zeroed pad rows of h2s, coeff is 0).
  {
    const int lane = tid & 31;
    const int wv   = tid >> 5;
    const int th   = wv >> 1, tw = wv & 1;
    const int nn   = lane & 15;          // M for A-frag, N for B-frag
    const int kb   = (lane >> 4) << 1;   // K pair base: 0 or 2
    v8f acc = {};
    #pragma unroll
    for (int dw = 0; dw < 3; ++dw) {
      float k0v = w2v[0 * 3 + dw];
      float k1v = w2v[1 * 3 + dw];
      float k2v = w2v[2 * 3 + dw];
      #pragma unroll
      for (int c5 = 0; c5 < 5; ++c5) {
        int k0 = 4 * c5 + kb;
        v2f A, B;
        A.x = tap3(k0     - nn, k0v, k1v, k2v);
        A.y = tap3(k0 + 1 - nn, k0v, k1v, k2v);
        const float* bp = &h2s[(16 * th + k0) * H2_S + 16 * tw + nn + dw];
        B.x = bp[0];
        B.y = bp[H2_S];
        // (neg_a, A, neg_b, B, c_mod, C, reuse_a, reuse_b)
        acc = __builtin_amdgcn_wmma_f32_16x16x4_f32(false, A, false, B,
                                                    (short)0, acc, false, false);
      }
    }
    // D layout: VGPR v holds M=v (lanes 0-15) / M=v+8 (lanes 16-31), N=lane&15
    const int Mb = (lane >> 4) * 8;
    #pragma unroll
    for (int v = 0; v < 8; ++v)
      h3s[(16 * th + Mb + v) * H3_S + 16 * tw + nn] = acc[v];
  }
  __syncthreads();

  // ---- Phase D: pool2 + store 16x16 outputs ----
  for (int idx = tid; idx < 256; idx += 128) {
    int p = idx >> 4, q = idx & 15;
    const float* hp = &h3s[(2 * p) * H3_S + 2 * q];
    float m = fmaxf(fmaxf(hp[0], hp[1]), fmaxf(hp[H3_S], hp[H3_S + 1]));
    out[((size_t)n * HOUT + (oh0 + p)) * HOUT + (ow0 + q)] = m;
  }
}

extern "C" void kernel_launch(void* const* d_in, const int* in_sizes, int n_in,
                              void* d_out, int out_size, void* d_ws, size_t ws_size,
                              hipStream_t stream) {
  (void)in_sizes; (void)n_in; (void)out_size; (void)d_ws; (void)ws_size;
  const float* x  = (const float*)d_in[0];
  const float* W1 = (const float*)d_in[1];
  const float* W2 = (const float*)d_in[2];
  float* out = (float*)d_out;
  // H=W=4096 fixed by the reference setup; 64x64 tiles of 16x16 outputs, N=2.
  dim3 grid(64, 64, 2), block(128);
  hipLaunchKernelGGL(fused_conv_pool_kernel, grid, block, 0, stream,
                     x, W1, W2, out);
}